// EncoderLinear_79748952752448
// MI455X (gfx1250) — compile-verified
//
#include <hip/hip_runtime.h>

typedef float v2f __attribute__((ext_vector_type(2)));
typedef float v8f __attribute__((ext_vector_type(8)));

#define HID 128

// Padded K extents (multiples of 8) for the three GEMMs
#define AU_KP 16    // real 13
#define AH_KP 136   // real 133
#define A3_KP 392   // real 386

// Workspace layout (floats):
//  aggU : [n_state][12]  : 0-1 sum pos_action[src], 2 sum dis, 3-10 sum u[src], 11 deg
//  aggH : [n_state][132] : 0-1 sum pos_state[src], 2 sum dis, 3-130 sum h[src], 131 deg
//  WTu  : [128][AU_KP]   : transposed, zero-padded W_u2h
//  WTh  : [128][AH_KP]   : transposed, zero-padded W_h2h
//  WTupd: [128][A3_KP]   : transposed, zero-padded W_upd

__global__ void zero_ws_kernel(float* ws, long n) {
  long i = (long)blockIdx.x * blockDim.x + threadIdx.x;
  long stride = (long)gridDim.x * blockDim.x;
  for (; i < n; i += stride) ws[i] = 0.0f;
}

// transpose + zero-pad the three weight matrices (tiny; runs each launch)
__global__ void pack_weights_kernel(const float* __restrict__ W_u2h,
                                    const float* __restrict__ W_h2h,
                                    const float* __restrict__ W_upd,
                                    float* __restrict__ WTu,
                                    float* __restrict__ WTh,
                                    float* __restrict__ WTupd) {
  const int stride = gridDim.x * blockDim.x;
  const int tid = blockIdx.x * blockDim.x + threadIdx.x;
  for (int i = tid; i < HID * AU_KP; i += stride) {
    int c = i / AU_KP, k = i % AU_KP;
    WTu[i] = (k < 13) ? W_u2h[k * HID + c] : 0.f;
  }
  for (int i = tid; i < HID * AH_KP; i += stride) {
    int c = i / AH_KP, k = i % AH_KP;
    WTh[i] = (k < 133) ? W_h2h[k * HID + c] : 0.f;
  }
  for (int i = tid; i < HID * A3_KP; i += stride) {
    int c = i / A3_KP, k = i % A3_KP;
    WTupd[i] = (k < 386) ? W_upd[k * HID + c] : 0.f;
  }
}

__global__ __launch_bounds__(256) void agg_a2s_kernel(
    const int* __restrict__ src, const int* __restrict__ dst,
    const float* __restrict__ dis, const float* __restrict__ pos_action,
    const float* __restrict__ u, float* __restrict__ aggU, int E) {
  int e = blockIdx.x * blockDim.x + threadIdx.x;
  if (e >= E) return;
  int s = src[e], d = dst[e];
  float* row = aggU + (long)d * 12;
  atomicAdd(row + 0, pos_action[2 * s]);
  atomicAdd(row + 1, pos_action[2 * s + 1]);
  atomicAdd(row + 2, dis[e]);
  const float* up = u + (long)s * 8;
#pragma unroll
  for (int i = 0; i < 8; ++i) atomicAdd(row + 3 + i, up[i]);
  atomicAdd(row + 11, 1.0f);
}

// one wave32 per edge: float4 gather of h[src], atomic scatter into aggH[dst]
__global__ __launch_bounds__(256) void agg_s2s_kernel(
    const int* __restrict__ src, const int* __restrict__ dst,
    const float* __restrict__ dis, const float* __restrict__ pos_state,
    const float* __restrict__ h, float* __restrict__ aggH, int E) {
  int gid = blockIdx.x * blockDim.x + threadIdx.x;
  int e = gid >> 5;
  int lane = gid & 31;
  if (e >= E) return;
  int s = src[e], d = dst[e];
  float* row = aggH + (long)d * 132;
  if (lane == 0) {
    atomicAdd(row + 0, pos_state[2 * s]);
    atomicAdd(row + 1, pos_state[2 * s + 1]);
    atomicAdd(row + 2, dis[e]);
    atomicAdd(row + 131, 1.0f);
  }
  const float4* hp4 = (const float4*)(h + (long)s * HID);
  float4 v = hp4[lane];  // 128 floats / 32 lanes = 1 float4 each
  float* p = row + 3 + 4 * lane;
  atomicAdd(p + 0, v.x);
  atomicAdd(p + 1, v.y);
  atomicAdd(p + 2, v.z);
  atomicAdd(p + 3, v.w);
}

// ---- fused node-update kernel: 16 nodes per block, 8 waves, WMMA f32 16x16x4 ----
#define A3_LD 393  // odd LDS strides -> bank-conflict-free A-fragment reads
#define AH_LD 137
#define AU_LD 17

__global__ __launch_bounds__(256, 1) void fused_node_kernel(
    const float* __restrict__ h, const float* __restrict__ pos_state,
    const float* __restrict__ aggU, const float* __restrict__ aggH,
    const float* __restrict__ WTu, const float* __restrict__ b_u2h,
    const float* __restrict__ WTh, const float* __restrict__ b_h2h,
    const float* __restrict__ WTupd, const float* __restrict__ b_upd,
    float* __restrict__ out, int n_state) {
  __shared__ float A[16 * A3_LD];   // stage-3 input rows [16 x 392(+pad)]
  __shared__ float Ah[16 * AH_LD];  // stage-2 input rows [16 x 133 -> 136]
  __shared__ float Au[16 * AU_LD];  // stage-1 input rows [16 x 13 -> 16]
  __shared__ float degU[16];
  __shared__ float degS[16];

  const int tid = threadIdx.x;
  const int n0 = blockIdx.x * 16;
  const int r = tid >> 4;   // row 0..15 for staging
  const int c0 = tid & 15;  // col group for staging
  const int n = n0 + r;
  const bool valid = (n < n_state);

  // ---- stage 0: assemble LDS input rows ----
  if (c0 == 0) {
    const float* au = aggU + (long)n * 12;
    const float* ah = aggH + (long)n * 132;
    float du = valid ? au[11] : 0.f;
    float ds = valid ? ah[131] : 0.f;
    degU[r] = du;
    degS[r] = ds;
    float p0 = valid ? pos_state[2 * n] : 0.f;
    float p1 = valid ? pos_state[2 * n + 1] : 0.f;
    float* AuR = Au + r * AU_LD;
    AuR[0] = valid ? au[0] : 0.f;  // sum pos_action
    AuR[1] = valid ? au[1] : 0.f;
    AuR[2] = du * p0;              // deg * pos_state[dst]
    AuR[3] = du * p1;
    AuR[4] = valid ? au[2] : 0.f;  // sum dis
#pragma unroll
    for (int i = 0; i < 8; ++i) AuR[5 + i] = valid ? au[3 + i] : 0.f;
    AuR[13] = AuR[14] = AuR[15] = 0.f;

    float* AhR = Ah + r * AH_LD;
    AhR[0] = valid ? ah[0] : 0.f;  // sum pos_state[src]
    AhR[1] = valid ? ah[1] : 0.f;
    AhR[2] = ds * p0;              // deg * pos_state[dst]
    AhR[3] = ds * p1;
    AhR[4] = valid ? ah[2] : 0.f;  // sum dis
    AhR[133] = AhR[134] = AhR[135] = 0.f;

    float* AR = A + r * A3_LD;
    AR[384] = p0;
    AR[385] = p1;
#pragma unroll
    for (int k = 386; k < A3_KP; ++k) AR[k] = 0.f;
  }
  {
    const float* hp = h + (long)n * HID;
    const float* ahp = aggH + (long)n * 132 + 3;
    for (int c = c0; c < HID; c += 16) {
      A[r * A3_LD + c] = valid ? hp[c] : 0.f;        // h -> cols 0..127
      Ah[r * AH_LD + 5 + c] = valid ? ahp[c] : 0.f;  // sum h[src]
    }
  }
  __syncthreads();

  // WMMA fragment coordinates (wave32; EXEC all-ones here, no divergence)
  const int lane = tid & 31;
  const int wv = tid >> 5;     // wave id -> 16-wide output column tile
  const int half = lane >> 4;  // 0: K,K+1 / rows 0-7 ; 1: K+2,K+3 / rows 8-15
  const int m = lane & 15;     // A row / B,C,D column within tile
  const int kb = half * 2;
  const int col = wv * 16 + m;  // absolute output column 0..127

  // ---- stage 1: sum_u tile = Au(16x13) @ W_u2h(13x128) + degU*b_u2h ----
  {
    const float* bp = WTu + col * AU_KP + kb;  // 8B-aligned (col*16, kb even)
    v8f acc = {};
#pragma unroll
    for (int k = 0; k < AU_KP; k += 4) {
      v2f a;
      a.x = Au[m * AU_LD + k + kb];
      a.y = Au[m * AU_LD + k + kb + 1];
      v2f b = *(const v2f*)(bp + k);
      acc = __builtin_amdgcn_wmma_f32_16x16x4_f32(false, a, false, b, (short)0,
                                                  acc, false, false);
    }
    float bu = b_u2h[col];
#pragma unroll
    for (int i = 0; i < 8; ++i) {
      int rr = i + 8 * half;
      A[rr * A3_LD + HID + col] = acc[i] + degU[rr] * bu;  // cols 128..255
    }
  }

  // ---- stage 2: mean_h tile = (Ah(16x133) @ W_h2h + degS*b_h2h)/max(degS,1) ----
  {
    const float* bp = WTh + col * AH_KP + kb;
    v8f acc = {};
#pragma unroll 2
    for (int k = 0; k < AH_KP; k += 4) {
      v2f a;
      a.x = Ah[m * AH_LD + k + kb];
      a.y = Ah[m * AH_LD + k + kb + 1];
      v2f b = *(const v2f*)(bp + k);
      acc = __builtin_amdgcn_wmma_f32_16x16x4_f32(false, a, false, b, (short)0,
                                                  acc, false, false);
    }
    float bh = b_h2h[col];
#pragma unroll
    for (int i = 0; i < 8; ++i) {
      int rr = i + 8 * half;
      float ds = degS[rr];
      A[rr * A3_LD + 2 * HID + col] =
          (acc[i] + ds * bh) / fmaxf(ds, 1.0f);  // cols 256..383
    }
  }
  __syncthreads();

  // ---- stage 3: out tile = A(16x386) @ W_upd(386x128) + b_upd ----
  {
    const float* bp = WTupd + col * A3_KP + kb;
    v8f acc = {};
#pragma unroll 2
    for (int k = 0; k < A3_KP; k += 4) {
      v2f a;
      a.x = A[m * A3_LD + k + kb];
      a.y = A[m * A3_LD + k + kb + 1];
      v2f b = *(const v2f*)(bp + k);
      acc = __builtin_amdgcn_wmma_f32_16x16x4_f32(false, a, false, b, (short)0,
                                                  acc, false, false);
    }
    float bo = b_upd[col];
#pragma unroll
    for (int i = 0; i < 8; ++i) {
      int rr = i + 8 * half;
      int nn = n0 + rr;
      if (nn < n_state) out[(long)nn * HID + col] = acc[i] + bo;
    }
  }
}

extern "C" void kernel_launch(void* const* d_in, const int* in_sizes, int n_in,
                              void* d_out, int out_size, void* d_ws,
                              size_t ws_size, hipStream_t stream) {
  const float* h = (const float*)d_in[0];
  const float* u = (const float*)d_in[1];
  const float* pos_state = (const float*)d_in[2];
  const float* pos_action = (const float*)d_in[3];
  const int* a2s_src = (const int*)d_in[4];
  const int* a2s_dst = (const int*)d_in[5];
  const float* a2s_dis = (const float*)d_in[6];
  const int* s2s_src = (const int*)d_in[7];
  const int* s2s_dst = (const int*)d_in[8];
  const float* s2s_dis = (const float*)d_in[9];
  const float* W_u2h = (const float*)d_in[10];
  const float* b_u2h = (const float*)d_in[11];
  const float* W_h2h = (const float*)d_in[12];
  const float* b_h2h = (const float*)d_in[13];
  const float* W_upd = (const float*)d_in[14];
  const float* b_upd = (const float*)d_in[15];
  float* out = (float*)d_out;

  const int n_state = in_sizes[2] / 2;
  const int E_a2s = in_sizes[4];
  const int E_s2s = in_sizes[7];

  float* aggU = (float*)d_ws;
  float* aggH = aggU + (size_t)n_state * 12;
  float* WTu = aggH + (size_t)n_state * 132;
  float* WTh = WTu + HID * AU_KP;
  float* WTupd = WTh + HID * AH_KP;
  const long agg_floats = (long)n_state * (12 + 132);

  zero_ws_kernel<<<2048, 256, 0, stream>>>(aggU, agg_floats);
  pack_weights_kernel<<<256, 256, 0, stream>>>(W_u2h, W_h2h, W_upd, WTu, WTh,
                                               WTupd);
  agg_a2s_kernel<<<(E_a2s + 255) / 256, 256, 0, stream>>>(
      a2s_src, a2s_dst, a2s_dis, pos_action, u, aggU, E_a2s);
  agg_s2s_kernel<<<(E_s2s + 7) / 8, 256, 0, stream>>>(
      s2s_src, s2s_dst, s2s_dis, pos_state, h, aggH, E_s2s);
  fused_node_kernel<<<(n_state + 15) / 16, 256, 0, stream>>>(
      h, pos_state, aggU, aggH, WTu, b_u2h, WTh, b_h2h, WTupd, b_upd, out,
      n_state);
}